// CausalScanMixer_2164663517289
// MI455X (gfx1250) — compile-verified
//
#include <hip/hip_runtime.h>

// ---------------------------------------------------------------------------
// CausalScanMixer for MI455X (gfx1250, wave32, WMMA)
//   out = x + causal @ W^T,  causal = EMA scan of x along S with d=sigmoid(p)
//   B=4, S=4096, D=1024
// Plan:
//   K1: chunked local scan (chunk L=128)  -> local[], carryEnd[]
//   K2: scan of chunk carries             -> stateIn[]  (tiny)
//   K3: FP32 WMMA GEMM, fixup causal = local + d^(i+1)*stateIn fused into
//       the A-tile LDS load; residual +x fused into epilogue.
// ---------------------------------------------------------------------------

typedef float v2f __attribute__((ext_vector_type(2)));
typedef float v4f __attribute__((ext_vector_type(4)));
typedef float v8f __attribute__((ext_vector_type(8)));

#define B_   4
#define S_   4096
#define D_   1024
#define L_   128                 // scan chunk length == GEMM M-tile height
#define NC_  (S_ / L_)           // 32 chunks per sequence
#define MTOT (B_ * S_)           // 16384 rows

__device__ __forceinline__ float dev_sigmoid(float v) {
  return 1.0f / (1.0f + __expf(-v));
}

// ---------------------------------------------------------------------------
// Kernel 1: per-chunk local EMA scan (zero initial state).
// Thread <-> (b, chunk, d-column); consecutive threads = consecutive d
// so every global access is a fully coalesced 128B wave transaction.
// ---------------------------------------------------------------------------
__global__ __launch_bounds__(256)
void scan_chunk_kernel(const float* __restrict__ x,
                       const float* __restrict__ decay_param,
                       float* __restrict__ local_scan,
                       float* __restrict__ carry_end) {
  const float dec = dev_sigmoid(decay_param[0]);
  const float w   = 1.0f - dec;

  const int gid = blockIdx.x * 256 + threadIdx.x;
  const int col = gid & (D_ - 1);
  const int bc  = gid >> 10;          // b*NC_ + chunk
  const int b   = bc >> 5;
  const int c   = bc & (NC_ - 1);

  const size_t base = ((size_t)(b * S_ + c * L_)) * D_ + col;

  float st = 0.0f;
#pragma unroll 4
  for (int i = 0; i < L_; ++i) {
    const size_t idx = base + (size_t)i * D_;
    // keep the read stream ahead of the serial FMA chain (global_prefetch_b8)
    if (i + 8 < L_) __builtin_prefetch(&x[idx + 8 * D_], 0, 1);
    const float xv = x[idx];
    st = fmaf(dec, st, w * xv);
    local_scan[idx] = st;
  }
  carry_end[(size_t)bc * D_ + col] = st;
}

// ---------------------------------------------------------------------------
// Kernel 2: exclusive scan of chunk carries per (b, d) column (32 steps).
//   stateIn[c] = causal value entering chunk c.
// ---------------------------------------------------------------------------
__global__ __launch_bounds__(256)
void carry_scan_kernel(const float* __restrict__ decay_param,
                       const float* __restrict__ carry_end,
                       float* __restrict__ state_in) {
  const float dec = dev_sigmoid(decay_param[0]);
  const float dL  = __expf((float)L_ * __logf(dec));   // d^128

  const int gid = blockIdx.x * 256 + threadIdx.x;      // 0..4095
  const int col = gid & (D_ - 1);
  const int b   = gid >> 10;

  float s = 0.0f;
#pragma unroll
  for (int c = 0; c < NC_; ++c) {
    const size_t idx = (size_t)(b * NC_ + c) * D_ + col;
    state_in[idx] = s;
    s = fmaf(dL, s, carry_end[idx]);
  }
}

// ---------------------------------------------------------------------------
// Kernel 3: out = x + causal @ W^T using V_WMMA_F32_16X16X4_F32.
//   256 threads = 8 waves; block tile 128(M) x 128(N); BK = 16.
//   Wave grid 4(M) x 2(N); wave tile 32 x 64 = 2x4 sub-tiles of 16x16.
//   A-tile load applies the scan fixup: a = local + d^(r+1) * stateIn.
//   M-tiles coincide with scan chunks (both 128 rows, b-major order), so
//   stateRow == mTile and r == row-within-chunk.
// ---------------------------------------------------------------------------
__global__ __launch_bounds__(256)
void gemm_wmma_kernel(const float* __restrict__ local_scan,
                      const float* __restrict__ state_in,
                      const float* __restrict__ Wg,
                      const float* __restrict__ x,
                      const float* __restrict__ decay_param,
                      float* __restrict__ out) {
  __shared__ float As[128][17];   // [m][k], pad to break bank conflicts
  __shared__ float Bs[128][17];   // [n][k] = W[nBase+n][k0+k]

  const int tid   = threadIdx.x;
  const int lane  = tid & 31;
  const int wave  = tid >> 5;
  const int waveM = wave & 3;     // 0..3
  const int waveN = wave >> 2;    // 0..1

  const int mTile = blockIdx.x >> 3;       // 0..127  (N innermost for L2 reuse)
  const int nTile = blockIdx.x & 7;        // 0..7
  const int mBase = mTile * 128;
  const int nBase = nTile * 128;

  const float dec  = dev_sigmoid(decay_param[0]);
  const float logd = __logf(dec);

  // each thread loads one A row / one B row, half of K (8 floats)
  const int   lr = tid >> 1;               // 0..127
  const int   kh = (tid & 1) * 8;          // 0 or 8
  const float pw = __expf((float)(lr + 1) * logd);   // d^(r+1)

  const float* __restrict__ stRow = state_in + (size_t)mTile * D_;

  v8f acc[2][4];
#pragma unroll
  for (int mt = 0; mt < 2; ++mt)
#pragma unroll
    for (int nt = 0; nt < 4; ++nt)
#pragma unroll
      for (int v = 0; v < 8; ++v)
        acc[mt][nt][v] = 0.0f;

  for (int k0 = 0; k0 < D_; k0 += 16) {
    // ---- stage A (with scan fixup) and B into LDS ----
    const float* ap = local_scan + (size_t)(mBase + lr) * D_ + (k0 + kh);
    const float* sp = stRow + (k0 + kh);
    const float* bp = Wg + (size_t)(nBase + lr) * D_ + (k0 + kh);

    const v4f a0 = *(const v4f*)ap;
    const v4f a1 = *(const v4f*)(ap + 4);
    const v4f s0 = *(const v4f*)sp;
    const v4f s1 = *(const v4f*)(sp + 4);
    const v4f b0 = *(const v4f*)bp;
    const v4f b1 = *(const v4f*)(bp + 4);

#pragma unroll
    for (int j = 0; j < 4; ++j) {
      As[lr][kh + j]     = fmaf(pw, s0[j], a0[j]);
      As[lr][kh + 4 + j] = fmaf(pw, s1[j], a1[j]);
      Bs[lr][kh + j]     = b0[j];
      Bs[lr][kh + 4 + j] = b1[j];
    }
    __syncthreads();

    // ---- 4 x (K=4) WMMA steps ----
    const int lh = lane & 15;
#pragma unroll
    for (int kk = 0; kk < 16; kk += 4) {
      // A 16x4 layout: lanes 0-15 hold K={0,1}, lanes 16-31 hold K={2,3}
      const int ks = kk + ((lane >> 4) << 1);

      v2f af[2];
#pragma unroll
      for (int mt = 0; mt < 2; ++mt) {
        const int row = waveM * 32 + mt * 16 + lh;
        v2f t; t.x = As[row][ks]; t.y = As[row][ks + 1];
        af[mt] = t;
      }
      v2f bf[4];
#pragma unroll
      for (int nt = 0; nt < 4; ++nt) {
        const int cc = waveN * 64 + nt * 16 + lh;
        v2f t; t.x = Bs[cc][ks]; t.y = Bs[cc][ks + 1];
        bf[nt] = t;
      }
#pragma unroll
      for (int mt = 0; mt < 2; ++mt)
#pragma unroll
        for (int nt = 0; nt < 4; ++nt)
          acc[mt][nt] = __builtin_amdgcn_wmma_f32_16x16x4_f32(
              /*neg_a=*/false, af[mt], /*neg_b=*/false, bf[nt],
              /*c_mod=*/(short)0, acc[mt][nt],
              /*reuse_a=*/false, /*reuse_b=*/false);
    }
    __syncthreads();
  }

  // ---- epilogue: out = x + acc (C/D layout: M = v + 8*(lane>=16), N = lane%16)
  const int lh  = lane & 15;
  const int mhi = (lane >> 4) * 8;
#pragma unroll
  for (int mt = 0; mt < 2; ++mt) {
#pragma unroll
    for (int nt = 0; nt < 4; ++nt) {
      const int gn  = nBase + waveN * 64 + nt * 16 + lh;
      const int gm0 = mBase + waveM * 32 + mt * 16 + mhi;
#pragma unroll
      for (int v = 0; v < 8; ++v) {
        const size_t o = (size_t)(gm0 + v) * D_ + gn;
        out[o] = x[o] + acc[mt][nt][v];
      }
    }
  }
}

// ---------------------------------------------------------------------------
// Launch. Workspace: local_scan (64 MiB) | carry_end (512 KiB) | state_in (512 KiB)
// ---------------------------------------------------------------------------
extern "C" void kernel_launch(void* const* d_in, const int* in_sizes, int n_in,
                              void* d_out, int out_size, void* d_ws, size_t ws_size,
                              hipStream_t stream) {
  const float* x     = (const float*)d_in[0];
  const float* decay = (const float*)d_in[1];
  const float* Wg    = (const float*)d_in[2];
  float* out = (float*)d_out;

  float* local_scan = (float*)d_ws;
  float* carry_end  = local_scan + (size_t)MTOT * D_;
  float* state_in   = carry_end + (size_t)B_ * NC_ * D_;

  scan_chunk_kernel<<<(B_ * NC_ * D_) / 256, 256, 0, stream>>>(
      x, decay, local_scan, carry_end);
  carry_scan_kernel<<<(B_ * D_) / 256, 256, 0, stream>>>(
      decay, carry_end, state_in);
  gemm_wmma_kernel<<<dim3(8 * (MTOT / 128)), 256, 0, stream>>>(
      local_scan, state_in, Wg, x, decay, out);
}